// SpectralFunctionWrapper_2783138808043
// MI455X (gfx1250) — compile-verified
//
#include <hip/hip_runtime.h>
#include <hip/hip_bf16.h>
#include <stdint.h>

typedef __attribute__((ext_vector_type(2))) float v2f;
typedef __attribute__((ext_vector_type(8))) float v8f;

#define NMAT 64        // matrices per block
#define RPAD 68        // tile row stride (floats): 64 + 4 pad, keeps 16B alignment for B128
#define GS   65        // gram/W/V stride: 64 + pad (stride%64==1 -> conflict free)
#define NSWEEP 6

__global__ __launch_bounds__(64)
void spectral_softplus_kernel(const float* __restrict__ x, float* __restrict__ out) {
  // phase1/2: tile[e][mat], e = c*8+k (128 rows of 64 floats, padded); phase5: W|V staging
  __shared__ float tile[128 * RPAD];
  __shared__ float grams[NMAT * GS];   // 8x8 gram per matrix

  const int tid  = threadIdx.x;        // 0..63, one matrix per thread for Jacobi
  const int lane = tid & 31;
  const int wave = tid >> 5;

  const int base_g = blockIdx.x * NMAT;      // first matrix id of this block
  const int bidx   = base_g >> 14;           // 16384 = H*W matrices per batch index
  const int hw     = base_g & 16383;

  // ------- Phase 1: async global -> LDS (CDNA5 GLOBAL_LOAD_ASYNC_TO_LDS_B128) -------
  // x row e (= c*8+k) of this block: 64 contiguous floats at ((b*128+e)<<14)+hw.
  // Each instruction moves 2 rows (16 lanes x 16B per row), per-lane LDS destinations.
  {
    const unsigned lds_tile = (unsigned)(uintptr_t)&tile[0];  // low 32b = LDS byte offset
    const int idx16 = lane & 15;                              // 16B chunk within a row
    const int rowh  = lane >> 4;                              // row within the pair
    #pragma unroll
    for (int u = 0; u < 32; ++u) {
      const int row = (wave << 6) + (u << 1) + rowh;          // 0..127
      const float* gp = x + (((size_t)(bidx * 128 + row)) << 14)
                          + (size_t)hw + (size_t)(idx16 << 2);
      const unsigned dst = lds_tile + (unsigned)((row * RPAD + (idx16 << 2)) << 2);
      asm volatile("global_load_async_to_lds_b128 %0, %1, off"
                   :: "v"(dst), "v"(gp) : "memory");
    }
    asm volatile("s_wait_asynccnt 0x0" ::: "memory");
  }
  __syncthreads();

  // ---------------- Phase 2: Gram via WMMA f32 16x16x4 ----------------
  // Two 8x16 matrices stacked into A (16x16); D = A*A^T; diagonal 8x8 blocks
  // are the two Gram matrices. B operand registers == A operand registers.
  {
    const int khalf  = lane >> 4;            // K-pair select within a K=4 step
    const int m      = lane & 15;            // stacked row (A) / col (B) index
    const int jm_off = m >> 3;
    const int r      = m & 7;
    #pragma unroll 1
    for (int t = 0; t < 16; ++t) {
      const int j0 = (wave << 5) + (t << 1);
      // element (mat j, row r, col c) lives at tile[(c*8+r)*RPAD + j]
      const float* src = &tile[r * RPAD + j0 + jm_off];
      v8f acc = {0.f, 0.f, 0.f, 0.f, 0.f, 0.f, 0.f, 0.f};
      #pragma unroll
      for (int s = 0; s < 4; ++s) {
        const int c0 = (s << 2) + (khalf << 1);
        v2f a;
        a.x = src[c0 * (8 * RPAD)];
        a.y = src[(c0 + 1) * (8 * RPAD)];
        acc = __builtin_amdgcn_wmma_f32_16x16x4_f32(
            false, a, false, a, (short)0, acc, false, false);
      }
      // C/D layout: VGPR r -> lanes 0-15: (M=r, N=lane); lanes 16-31: (M=r+8, N=lane-16)
      const bool lo = lane < 8;
      const bool hi = lane >= 24;
      if (lo | hi) {
        const int gj  = lo ? j0 : (j0 + 1);
        const int col = lo ? lane : (lane - 24);
        float* gd = &grams[gj * GS + col];
        #pragma unroll
        for (int rr = 0; rr < 8; ++rr) gd[rr * 8] = acc[rr];
      }
    }
  }
  __syncthreads();

  // ---------------- Phase 3: per-lane 8x8 cyclic Jacobi ----------------
  float Am[8][8], Vm[8][8];
  {
    const float* gsrc = &grams[tid * GS];
    #pragma unroll
    for (int i = 0; i < 8; ++i)
      #pragma unroll
      for (int j = 0; j < 8; ++j) {
        Am[i][j] = gsrc[i * 8 + j];
        Vm[i][j] = (i == j) ? 1.0f : 0.0f;
      }
  }
  for (int sweep = 0; sweep < NSWEEP; ++sweep) {
    #pragma unroll
    for (int p = 0; p < 7; ++p) {
      #pragma unroll
      for (int q = p + 1; q < 8; ++q) {
        const float apq = Am[p][q];
        float denom = 2.0f * apq;
        denom = (denom == 0.0f) ? 1e-30f : denom;
        const float tau = (Am[q][q] - Am[p][p]) / denom;
        const float tt  = copysignf(1.0f, tau) /
                          (fabsf(tau) + sqrtf(fmaf(tau, tau, 1.0f)));
        const float c   = 1.0f / sqrtf(fmaf(tt, tt, 1.0f));
        const float s   = tt * c;
        #pragma unroll
        for (int k = 0; k < 8; ++k) {         // A <- A*G (columns p,q)
          const float akp = Am[k][p], akq = Am[k][q];
          Am[k][p] = fmaf(c, akp, -s * akq);
          Am[k][q] = fmaf(s, akp,  c * akq);
        }
        #pragma unroll
        for (int k = 0; k < 8; ++k) {         // A <- G^T*A (rows p,q)
          const float apk = Am[p][k], aqk = Am[q][k];
          Am[p][k] = fmaf(c, apk, -s * aqk);
          Am[q][k] = fmaf(s, apk,  c * aqk);
        }
        #pragma unroll
        for (int k = 0; k < 8; ++k) {         // V <- V*G
          const float vkp = Vm[k][p], vkq = Vm[k][q];
          Vm[k][p] = fmaf(c, vkp, -s * vkq);
          Vm[k][q] = fmaf(s, vkp,  c * vkq);
        }
      }
    }
  }

  // ---------------- Phase 4: spectral map s = softplus(sqrt(max(l,eps))) ----------------
  float sd[8];
  #pragma unroll
  for (int k = 0; k < 8; ++k) {
    const float lam = fmaxf(Am[k][k], 1e-12f);
    const float r   = sqrtf(lam);
    const float sp  = log1pf(__expf(r));
    sd[k] = (r > 20.0f) ? r : sp;             // overflow-safe softplus
  }

  // Stage W = V*diag(s) and V into LDS (reuse tile: 2*NMAT*GS = 8320 <= 128*RPAD = 8704)
  float* wbuf = tile;
  float* vbuf = tile + NMAT * GS;
  #pragma unroll
  for (int i = 0; i < 8; ++i)
    #pragma unroll
    for (int k = 0; k < 8; ++k) {
      wbuf[tid * GS + i * 8 + k] = Vm[i][k] * sd[k];
      vbuf[tid * GS + i * 8 + k] = Vm[i][k];
    }
  __syncthreads();

  // ---------------- Phase 5: reconstruct D = W * V^T via WMMA (K=8) ----------------
  {
    const int khalf = lane >> 4;
    const int m     = lane & 15;
    #pragma unroll 1
    for (int t = 0; t < 16; ++t) {
      const int j0 = (wave << 5) + (t << 1);
      const int jm = j0 + (m >> 3);
      const float* wsrc = &wbuf[jm * GS + (m & 7) * 8]; // A row m: W[m&7][0..7]
      const float* vsrc = &vbuf[jm * GS + (m & 7) * 8]; // B col n=m: V[n&7][0..7]
      v8f acc = {0.f, 0.f, 0.f, 0.f, 0.f, 0.f, 0.f, 0.f};
      #pragma unroll
      for (int s = 0; s < 2; ++s) {
        const int c0 = (s << 2) + (khalf << 1);
        v2f a; a.x = wsrc[c0]; a.y = wsrc[c0 + 1];
        v2f b; b.x = vsrc[c0]; b.y = vsrc[c0 + 1];
        acc = __builtin_amdgcn_wmma_f32_16x16x4_f32(
            false, a, false, b, (short)0, acc, false, false);
      }
      const bool lo = lane < 8;
      const bool hi = lane >= 24;
      if (lo | hi) {
        const int gj  = lo ? j0 : (j0 + 1);
        const int col = lo ? lane : (lane - 24);
        float* od = out + (size_t)(base_g + gj) * 64 + col;
        #pragma unroll
        for (int r = 0; r < 8; ++r) od[r * 8] = acc[r];
      }
    }
  }
}

extern "C" void kernel_launch(void* const* d_in, const int* in_sizes, int n_in,
                              void* d_out, int out_size, void* d_ws, size_t ws_size,
                              hipStream_t stream) {
  (void)in_sizes; (void)n_in; (void)out_size; (void)d_ws; (void)ws_size;
  const float* x = (const float*)d_in[0];
  float* out = (float*)d_out;
  const int n_mat = 8 * 128 * 128;                 // B*H*W = 131072
  dim3 grid(n_mat / NMAT), block(64);
  spectral_softplus_kernel<<<grid, block, 0, stream>>>(x, out);
}